// CCA_SSG_68229850464275
// MI455X (gfx1250) — compile-verified
//
#include <hip/hip_runtime.h>
#include <hip/hip_bf16.h>

#define NNODES 100000
#define NEDGES 800000
#define DIM 128
#define NM ((size_t)NNODES * DIM)

typedef float v2f __attribute__((ext_vector_type(2)));
typedef float v8f __attribute__((ext_vector_type(8)));

// ---------------------------------------------------------------- utilities
__global__ __launch_bounds__(256) void k_zero(float* __restrict__ p, size_t n) {
  size_t i = (size_t)blockIdx.x * blockDim.x + threadIdx.x;
  size_t stride = (size_t)gridDim.x * blockDim.x;
  for (; i < n; i += stride) p[i] = 0.0f;
}

__global__ __launch_bounds__(256) void k_deg(const int* __restrict__ idx,
                                             float* __restrict__ deg, int n) {
  int e = blockIdx.x * blockDim.x + threadIdx.x;
  if (e < n) atomicAdd(&deg[idx[e]], 1.0f);
}

__global__ __launch_bounds__(256) void k_invsqrt(float* __restrict__ d, int n) {
  int i = blockIdx.x * blockDim.x + threadIdx.x;
  if (i < n) d[i] = rsqrtf(fmaxf(d[i], 1.0f));
}

// ------------------------------------------------- fused scale + GEMM (WMMA)
// Y[M,128] = (X * rowscale[:,None]) @ W     using V_WMMA_F32_16X16X4_F32
// One block = 8 wave32s: 16 rows of X staged in LDS (scaled), each wave owns a
// 16-wide N stripe, K=128 -> 32 chained wmma ops per wave.
__global__ __launch_bounds__(256) void k_gemm_scaled(const float* __restrict__ X,
                                                     const float* __restrict__ W,
                                                     const float* __restrict__ rowscale,
                                                     float* __restrict__ Y) {
  __shared__ float As[16 * 132];                 // stride 132 -> no bank conflicts
  const int m0 = blockIdx.x * 16;
#pragma unroll
  for (int j = 0; j < 8; ++j) {                  // 2048 floats / 256 threads
    int i = threadIdx.x + 256 * j;
    int r = i >> 7, c = i & 127;
    As[r * 132 + c] = X[(size_t)(m0 + r) * DIM + c] * rowscale[m0 + r];
  }
  __syncthreads();

  const int lane = threadIdx.x & 31;
  const int l    = lane & 15;                    // row (A) / col (B,D)
  const int half = lane >> 4;                    // K-pair select
  const int nb   = (threadIdx.x >> 5) * 16;      // N stripe per wave
  const float* a_row = &As[l * 132];

  v8f acc = {0.f, 0.f, 0.f, 0.f, 0.f, 0.f, 0.f, 0.f};
#pragma unroll
  for (int k0 = 0; k0 < 128; k0 += 4) {
    const int ka = k0 + 2 * half;
    v2f a, b;
    a[0] = a_row[ka];
    a[1] = a_row[ka + 1];
    b[0] = W[(size_t)ka * DIM + nb + l];
    b[1] = W[(size_t)(ka + 1) * DIM + nb + l];
    acc = __builtin_amdgcn_wmma_f32_16x16x4_f32(false, a, false, b,
                                                (short)0, acc, false, false);
  }
  // D layout: VGPR v -> row (v + 8*half), col nb+l
  float* yp = Y + (size_t)(m0 + 8 * half) * DIM + nb + l;
#pragma unroll
  for (int v = 0; v < 8; ++v) yp[(size_t)v * DIM] = acc[v];
}

// --------------------------------------------------- SpMM gather/scatter-add
// One wave per edge; 32 lanes x float4 = 128 features per edge.
__global__ __launch_bounds__(256) void k_scatter(const float* __restrict__ Xw,
                                                 const int* __restrict__ src,
                                                 const int* __restrict__ dst,
                                                 float* __restrict__ agg, int n_edges) {
  const int lane = threadIdx.x & 31;
  const int wave = (blockIdx.x << 3) + (threadIdx.x >> 5);
  const int nw   = gridDim.x << 3;
  for (int e = wave; e < n_edges; e += nw) {
    __builtin_prefetch(&src[e + nw], 0, 0);      // -> global_prefetch_b8
    __builtin_prefetch(&dst[e + nw], 0, 0);
    const int s = src[e];
    const int d = dst[e];
    const float4 v = ((const float4*)(Xw + (size_t)s * DIM))[lane];
    float* o = agg + (size_t)d * DIM + (lane << 2);
    atomicAdd(o + 0, v.x);
    atomicAdd(o + 1, v.y);
    atomicAdd(o + 2, v.z);
    atomicAdd(o + 3, v.w);
  }
}

// ------------------------------------------------ in-scale + bias (+ ReLU)
__global__ __launch_bounds__(256) void k_finalize(const float* __restrict__ agg,
                                                  const float* __restrict__ iscale,
                                                  const float* __restrict__ bias,
                                                  float* __restrict__ out, int relu) {
  size_t i = (size_t)blockIdx.x * blockDim.x + threadIdx.x;
  if (i >= NM) return;
  int row = (int)(i >> 7), c = (int)(i & 127);
  float v = agg[i] * iscale[row] + bias[c];
  if (relu) v = fmaxf(v, 0.0f);
  out[i] = v;
}

// ------------------------------------------------------- column statistics
__global__ __launch_bounds__(256) void k_colstats(const float* __restrict__ h,
                                                  float* __restrict__ sums,
                                                  float* __restrict__ sumsq, int M) {
  __shared__ float ss[128], sq[128];
  const int c = threadIdx.x & 127;
  const int r0 = threadIdx.x >> 7;
  const int base = blockIdx.x * 256;
  const int end = min(M, base + 256);
  float s = 0.f, q = 0.f;
  for (int r = base + r0; r < end; r += 2) {
    float v = h[(size_t)r * DIM + c];
    s += v;
    q += v * v;
  }
  if (r0 == 1) { ss[c] = s; sq[c] = q; }
  __syncthreads();
  if (r0 == 0) {
    atomicAdd(&sums[c], s + ss[c]);
    atomicAdd(&sumsq[c], q + sq[c]);
  }
}

__global__ void k_statsfin(const float* __restrict__ sums,
                           const float* __restrict__ sumsq,
                           float* __restrict__ mean, float* __restrict__ invstd, int M) {
  int c = threadIdx.x;
  float mu = sums[c] / (float)M;
  float var = (sumsq[c] - sums[c] * mu) / (float)(M - 1);   // ddof=1
  mean[c] = mu;
  invstd[c] = rsqrtf(fmaxf(var, 1e-30f));
}

__global__ __launch_bounds__(256) void k_zscore(float* __restrict__ z,
                                                const float* __restrict__ mean,
                                                const float* __restrict__ invstd) {
  size_t i = (size_t)blockIdx.x * blockDim.x + threadIdx.x;
  if (i >= NM) return;
  int c = (int)(i & 127);
  z[i] = (z[i] - mean[c]) * invstd[c];
}

// --------------------------------------------------------------------------
extern "C" void kernel_launch(void* const* d_in, const int* in_sizes, int n_in,
                              void* d_out, int out_size, void* d_ws, size_t ws_size,
                              hipStream_t stream) {
  (void)in_sizes; (void)n_in; (void)out_size; (void)ws_size;
  const float* feat1 = (const float*)d_in[0];
  const float* feat2 = (const float*)d_in[1];
  const float* W1    = (const float*)d_in[2];
  const float* b1    = (const float*)d_in[3];
  const float* W2    = (const float*)d_in[4];
  const float* b2    = (const float*)d_in[5];
  const int*   src1  = (const int*)d_in[6];
  const int*   dst1  = (const int*)d_in[7];
  const int*   src2  = (const int*)d_in[8];
  const int*   dst2  = (const int*)d_in[9];
  float* out = (float*)d_out;

  // workspace layout
  float* bufA   = (float*)d_ws;           // 12.8M floats
  float* bufB   = bufA + NM;              // 12.8M floats
  float* dso    = bufB + NM;              // 100k (deg_out^-1/2)
  float* dsi    = dso + NNODES;           // 100k (deg_in^-1/2)
  float* sums   = dsi + NNODES;           // 128
  float* sumsq  = sums + 128;             // 128
  float* meanv  = sumsq + 128;            // 128
  float* invstd = meanv + 128;            // 128

  const int B_EDGE = (NEDGES + 255) / 256;    // 3125
  const int B_NODE = (NNODES + 255) / 256;    // 391
  const int B_NM   = (int)((NM + 255) / 256); // 50000
  const int B_GEMM = NNODES / 16;             // 6250 (exact)

  auto run_graph = [&](const float* feat, const int* src, const int* dst, float* zout) {
    // degree normalizers (shared by both layers)
    k_zero<<<B_NODE, 256, 0, stream>>>(dso, NNODES);
    k_zero<<<B_NODE, 256, 0, stream>>>(dsi, NNODES);
    k_deg<<<B_EDGE, 256, 0, stream>>>(src, dso, NEDGES);
    k_deg<<<B_EDGE, 256, 0, stream>>>(dst, dsi, NEDGES);
    k_invsqrt<<<B_NODE, 256, 0, stream>>>(dso, NNODES);
    k_invsqrt<<<B_NODE, 256, 0, stream>>>(dsi, NNODES);

    // layer 1: bufA = (feat*dso)@W1 ; bufB = scatter ; bufA = relu(bufB*dsi + b1)
    k_gemm_scaled<<<B_GEMM, 256, 0, stream>>>(feat, W1, dso, bufA);
    k_zero<<<B_NM, 256, 0, stream>>>(bufB, NM);
    k_scatter<<<4096, 256, 0, stream>>>(bufA, src, dst, bufB, NEDGES);
    k_finalize<<<B_NM, 256, 0, stream>>>(bufB, dsi, b1, bufA, 1);

    // layer 2: bufB = (bufA*dso)@W2 ; bufA = scatter ; zout = bufA*dsi + b2
    k_gemm_scaled<<<B_GEMM, 256, 0, stream>>>(bufA, W2, dso, bufB);
    k_zero<<<B_NM, 256, 0, stream>>>(bufA, NM);
    k_scatter<<<4096, 256, 0, stream>>>(bufB, src, dst, bufA, NEDGES);
    k_finalize<<<B_NM, 256, 0, stream>>>(bufA, dsi, b2, zout, 0);

    // per-column z-score (in place in d_out)
    k_zero<<<1, 256, 0, stream>>>(sums, 256);   // sums + sumsq contiguous
    k_colstats<<<B_NODE, 256, 0, stream>>>(zout, sums, sumsq, NNODES);
    k_statsfin<<<1, 128, 0, stream>>>(sums, sumsq, meanv, invstd, NNODES);
    k_zscore<<<B_NM, 256, 0, stream>>>(zout, meanv, invstd);
  };

  run_graph(feat1, src1, dst1, out);
  run_graph(feat2, src2, dst2, out + NM);
}